// GIoU_34308198760690
// MI455X (gfx1250) — compile-verified
//
#include <hip/hip_runtime.h>
#include <hip/hip_bf16.h>

#define GIOU_EPS 1e-8f

typedef float v2f __attribute__((ext_vector_type(2)));
typedef float v8f __attribute__((ext_vector_type(8)));

#define TPB    256
#define ITEMS  4
#define LSLOTS 17   // float2 per thread: bufA[0..7], bufB[8..15], pad[16] (odd skew)

// ---------------------------------------------------------------------------
// Sutherland-Hodgman clip, LDS -> LDS. src has n valid float2 vertices.
// Returns emitted count with the reference's "<3 -> discard" rule applied.
// ---------------------------------------------------------------------------
__device__ __forceinline__ int clip_lds(const float2* __restrict__ src, int n,
                                        float ax, float ay, float ex, float ey,
                                        float2* __restrict__ dst) {
    int m = 0;
    if (n > 0) {
        float2 prev = src[n - 1];
        float sprev = ex * (prev.y - ay) - ey * (prev.x - ax);
        for (int i = 0; i < n; ++i) {
            const float2 cur = src[i];
            const float scur = ex * (cur.y - ay) - ey * (cur.x - ax);
            const bool cin = (scur >= 0.0f);
            const bool pin = (sprev >= 0.0f);
            if (pin != cin) {
                const float ddx = cur.x - prev.x, ddy = cur.y - prev.y;
                const float t = ((ax - prev.x) * ey - (ay - prev.y) * ex) /
                                (ddx * ey - ddy * ex + GIOU_EPS);
                dst[m] = make_float2(prev.x + t * ddx, prev.y + t * ddy);
                ++m;
            }
            if (cin) { dst[m] = cur; ++m; }
            prev = cur; sprev = scur;
        }
    }
    return (m < 3) ? 0 : m;
}

// ---------------------------------------------------------------------------
// Per-element (1 - giou). bufA/bufB: this thread's private LDS scratch.
// ---------------------------------------------------------------------------
__device__ __forceinline__ float giou_one(const float* __restrict__ bp,
                                          const float* __restrict__ bt,
                                          float2* __restrict__ bufA,
                                          float2* __restrict__ bufB) {
    const float cx1 = bp[0], cy1 = bp[1];
    const float w1  = __expf(bp[3]), l1 = __expf(bp[4]);
    const float yaw1 = bp[6];
    const float cx2 = bt[0], cy2 = bt[1];
    const float w2  = __expf(bt[3]), l2 = __expf(bt[4]);
    const float yaw2 = bt[6];

    const float s1 = __sinf(yaw1), c1 = __cosf(yaw1);
    const float s2 = __sinf(yaw2), c2 = __cosf(yaw2);

    float v1x[4], v1y[4], v2x[4], v2y[4];
    {
        const float dx = 0.5f * w1, dy = 0.5f * l1;
        const float rx[4] = {-dx, dx, dx, -dx};
        const float ry[4] = {-dy, -dy, dy, dy};
#pragma unroll
        for (int k = 0; k < 4; ++k) {
            v1x[k] = rx[k] * c1 - ry[k] * s1 + cx1;
            v1y[k] = rx[k] * s1 + ry[k] * c1 + cy1;
        }
    }
    {
        const float dx = 0.5f * w2, dy = 0.5f * l2;
        const float rx[4] = {-dx, dx, dx, -dx};
        const float ry[4] = {-dy, -dy, dy, dy};
#pragma unroll
        for (int k = 0; k < 4; ++k) {
            v2x[k] = rx[k] * c2 - ry[k] * s2 + cx2;
            v2y[k] = rx[k] * s2 + ry[k] * c2 + cy2;
        }
    }

    // ---- clip 1: quad1 (registers, n == 4 constant) by edge v2[0]->v2[1] ----
    int n1 = 0;
    {
        const float ax = v2x[0], ay = v2y[0];
        const float ex = v2x[1] - ax, ey = v2y[1] - ay;
        float prx = v1x[3], pry = v1y[3];
        float sprev = ex * (pry - ay) - ey * (prx - ax);
#pragma unroll
        for (int i = 0; i < 4; ++i) {
            const float cxr = v1x[i], cyr = v1y[i];
            const float scur = ex * (cyr - ay) - ey * (cxr - ax);
            const bool cin = (scur >= 0.0f);
            const bool pin = (sprev >= 0.0f);
            if (pin != cin) {
                const float ddx = cxr - prx, ddy = cyr - pry;
                const float t = ((ax - prx) * ey - (ay - pry) * ex) /
                                (ddx * ey - ddy * ex + GIOU_EPS);
                bufA[n1] = make_float2(prx + t * ddx, pry + t * ddy);
                ++n1;
            }
            if (cin) { bufA[n1] = make_float2(cxr, cyr); ++n1; }
            prx = cxr; pry = cyr; sprev = scur;
        }
        if (n1 < 3) n1 = 0;
    }

    // ---- clips 2,3: LDS ping-pong ----
    const int n2 = clip_lds(bufA, n1, v2x[1], v2y[1],
                            v2x[2] - v2x[1], v2y[2] - v2y[1], bufB);
    const int n3 = clip_lds(bufB, n2, v2x[2], v2y[2],
                            v2x[3] - v2x[2], v2y[3] - v2y[2], bufA);

    // ---- clip 4 fused with shoelace accumulation (reference emit order) ----
    float cs = 0.0f;
    int m4 = 0;
    {
        const float ax = v2x[3], ay = v2y[3];
        const float ex = v2x[0] - ax, ey = v2y[0] - ay;
        float fx = 0.f, fy = 0.f, px = 0.f, py = 0.f;
        bool hasv = false;
        if (n3 > 0) {
            float2 prev = bufA[n3 - 1];
            float sprev = ex * (prev.y - ay) - ey * (prev.x - ax);
            for (int i = 0; i < n3; ++i) {
                const float2 cur = bufA[i];
                const float scur = ex * (cur.y - ay) - ey * (cur.x - ax);
                const bool cin = (scur >= 0.0f);
                const bool pin = (sprev >= 0.0f);
                if (pin != cin) {
                    const float ddx = cur.x - prev.x, ddy = cur.y - prev.y;
                    const float t = ((ax - prev.x) * ey - (ay - prev.y) * ex) /
                                    (ddx * ey - ddy * ex + GIOU_EPS);
                    const float ix = prev.x + t * ddx, iy = prev.y + t * ddy;
                    if (hasv) cs += px * iy - py * ix;
                    else { fx = ix; fy = iy; hasv = true; }
                    px = ix; py = iy; ++m4;
                }
                if (cin) {
                    if (hasv) cs += px * cur.y - py * cur.x;
                    else { fx = cur.x; fy = cur.y; hasv = true; }
                    px = cur.x; py = cur.y; ++m4;
                }
                prev = cur; sprev = scur;
            }
        }
        if (hasv) cs += px * fy - py * fx;   // wrap term
    }
    const bool ok = (n1 >= 3) & (n2 >= 3) & (n3 >= 3) & (m4 >= 3);
    const float inter = ok ? fabsf(cs * 0.5f) : 0.0f;

    // Axis-aligned enclosure over all 8 original corners.
    float xmin = v1x[0], xmax = v1x[0], ymin = v1y[0], ymax = v1y[0];
#pragma unroll
    for (int k = 1; k < 4; ++k) {
        xmin = fminf(xmin, v1x[k]); xmax = fmaxf(xmax, v1x[k]);
        ymin = fminf(ymin, v1y[k]); ymax = fmaxf(ymax, v1y[k]);
    }
#pragma unroll
    for (int k = 0; k < 4; ++k) {
        xmin = fminf(xmin, v2x[k]); xmax = fmaxf(xmax, v2x[k]);
        ymin = fminf(ymin, v2y[k]); ymax = fmaxf(ymax, v2y[k]);
    }
    const float enc = (xmax - xmin) * (ymax - ymin);

    const float area1 = w1 * l1;
    const float area2 = w2 * l2;
    const float uni = area1 + area2 - inter;
    const float giou = inter / (uni + GIOU_EPS) - (enc - uni) / (enc + GIOU_EPS);
    return 1.0f - giou;
}

// ---------------------------------------------------------------------------
// Pass 1: per-block deterministic partial sums. Each thread owns a skewed
// 17-float2 LDS region (bank-pair (17*tid+i) mod 32 is conflict-free for
// uniform i under wave32). ITEMS loop not unrolled: one copy of clip code.
// ---------------------------------------------------------------------------
__global__ void giou_partial_kernel(const float* __restrict__ box,
                                    const float* __restrict__ tgt,
                                    float* __restrict__ partials, int N) {
    __shared__ float2 poly[TPB * LSLOTS];
    float2* bufA = poly + threadIdx.x * LSLOTS;
    float2* bufB = bufA + 8;

    const int base = blockIdx.x * (TPB * ITEMS);
    float acc = 0.0f;

#pragma unroll 1
    for (int it = 0; it < ITEMS; ++it) {
        const int i = base + it * TPB + threadIdx.x;
        if (i < N) {
            // gfx1250 prefetch (global_prefetch_b8) for the next tile's rows.
            if (it + 1 < ITEMS && i + TPB < N) {
                __builtin_prefetch(box + (size_t)(i + TPB) * 7, 0, 0);
                __builtin_prefetch(tgt + (size_t)(i + TPB) * 7, 0, 0);
            }
            acc += giou_one(box + (size_t)i * 7, tgt + (size_t)i * 7, bufA, bufB);
        }
    }

    // wave32 reduce
    for (int off = 16; off > 0; off >>= 1) acc += __shfl_down(acc, off, 32);

    __shared__ float sdata[TPB / 32];
    const int lane = threadIdx.x & 31;
    const int wave = threadIdx.x >> 5;
    if (lane == 0) sdata[wave] = acc;
    __syncthreads();

    if (wave == 0) {
        float v = (lane < (TPB / 32)) ? sdata[lane] : 0.0f;
        for (int off = 4; off > 0; off >>= 1) v += __shfl_down(v, off, 32);
        if (lane == 0) partials[blockIdx.x] = v;
    }
}

// ---------------------------------------------------------------------------
// Pass 2: one wave32 reduces all block partials. The 64 lane-accumulators are
// packed into the B operand of v_wmma_f32_16x16x4_f32 with A = ones(16x4):
//   D[m][n] = sum_k B[k][n]  ->  sum over N of D row 0 == sum of all 64 slots.
// Deterministic (fixed lane->slot mapping), exercises the CDNA5 WMMA path.
// ---------------------------------------------------------------------------
__global__ void giou_final_kernel(const float* __restrict__ partials, int P,
                                  const int* __restrict__ avg_factor,
                                  float* __restrict__ out) {
    const int lane = threadIdx.x;  // 32 threads, one wave
    float b0 = 0.0f, b1 = 0.0f;
    for (int j = lane; j < P; j += 64) b0 += partials[j];
    for (int j = lane + 32; j < P; j += 64) b1 += partials[j];

    v2f a; a.x = 1.0f; a.y = 1.0f;         // A = ones(16x4)
    v2f b; b.x = b0;   b.y = b1;           // B holds the 64 partial sums
    v8f c = {};
    v8f d = __builtin_amdgcn_wmma_f32_16x16x4_f32(
        /*neg_a=*/false, a, /*neg_b=*/false, b,
        /*c_mod=*/(short)0, c, /*reuse_a=*/false, /*reuse_b=*/false);

    // D VGPR0: lanes 0..15 = colsum[N=lane] (lanes 16..31 duplicate M=8 row).
    float v = d[0];
    v += __shfl_xor(v, 8, 32);
    v += __shfl_xor(v, 4, 32);
    v += __shfl_xor(v, 2, 32);
    v += __shfl_xor(v, 1, 32);

    if (lane == 0) {
        const float denom = fmaxf((float)avg_factor[0], 1.0f);
        out[0] = v / denom;
    }
}

// ---------------------------------------------------------------------------
extern "C" void kernel_launch(void* const* d_in, const int* in_sizes, int n_in,
                              void* d_out, int out_size, void* d_ws, size_t ws_size,
                              hipStream_t stream) {
    const float* box = (const float*)d_in[0];
    const float* tgt = (const float*)d_in[1];
    const int*   af  = (const int*)d_in[2];

    const int N = in_sizes[0] / 7;
    const int TILE = TPB * ITEMS;
    const int blocks = (N + TILE - 1) / TILE;

    float* partials = (float*)d_ws;

    giou_partial_kernel<<<blocks, TPB, 0, stream>>>(box, tgt, partials, N);
    giou_final_kernel<<<1, 32, 0, stream>>>(partials, blocks, af, (float*)d_out);
}